// FACoef_15633680958284
// MI455X (gfx1250) — compile-verified
//
#include <hip/hip_runtime.h>
#include <stdint.h>

// FACoef: out[b] = sum_{i,j} coef[i,j] * (sum(x_b^(i+2)))^(j+1) / n^(i+j+2)
// sum(A^k) = 1^T A^k 1  ->  matvec chain (4 matvecs/sample), no matrix powers.
// HBM-bound: 256 MB once @ 23.3 TB/s ~= 11 us floor.
// One workgroup (8 wave32s) per sample:
//   - TENSOR_LOAD_TO_LDS (TDM) DMAs the 64 KB tile into LDS, hardware-padding
//     the row pitch 128 -> 132 dwords (bank-conflict-free ds_load_b64).
//   - Matvec chain via V_WMMA_F32_16X16X4_F32 (f32 required: sums reach ~2e9).

#define USE_TDM 1

typedef float v2f __attribute__((ext_vector_type(2)));
typedef float v8f __attribute__((ext_vector_type(8)));
typedef uint32_t u32x4 __attribute__((ext_vector_type(4)));
typedef uint32_t u32x8 __attribute__((ext_vector_type(8)));

#define GSZ   128
#define PAD   132          // 132 % 64 == 4 -> conflict-free b64 LDS loads
#define NPOW  4            // r1..r4 (r0 = ones)

__global__ __launch_bounds__(256)
void facoef_kernel(const float* __restrict__ x,
                   const float* __restrict__ coef,
                   float* __restrict__ out)
{
    __shared__ float xs[GSZ * PAD];          // 67,584 B: padded 128x128 tile
    __shared__ float rall[NPOW + 1][GSZ];    // r0..r4 vectors

    const int bid = blockIdx.x;
    const int tid = threadIdx.x;
    const float* __restrict__ xg = x + (size_t)bid * (GSZ * GSZ);

#if USE_TDM
    // ---- TDM: one tensor_load_to_lds DMAs the whole padded tile ----
    if (tid < 32) {                       // single wave issues the DMA
        const uint64_t ga  = (uint64_t)(uintptr_t)xg;     // tile == tensor start
        const uint32_t lds = (uint32_t)(uintptr_t)&xs[0]; // LDS byte offset

        u32x4 g0;
        g0[0] = 1u;                                   // count=1 (valid user D#)
        g0[1] = lds;                                  // lds_addr[31:0]
        g0[2] = (uint32_t)ga;                         // global_addr[31:0]
        g0[3] = ((uint32_t)(ga >> 32) & 0x01FFFFFFu)  // global_addr[56:32]
                | (2u << 30);                         // type=2 ("image")

        u32x8 g1;
        g1[0] = (2u << 16)        // data_size=2 -> 4-byte elements
              | (1u << 20)       // pad_enable
              | (6u << 22)       // pad_interval: every 128 dwords
              | (3u << 25);      // pad_amount: 4 dwords  (pitch 128->132)
        g1[1] = (128u << 16);    // tensor_dim0[15:0]=128 (atomic_barrier_addr=0)
        g1[2] = (128u << 16);    // tensor_dim0[31:16]=0 | tensor_dim1[15:0]=128
        g1[3] = (128u << 16);    // tensor_dim1[31:16]=0 | tile_dim0=128
        g1[4] = 128u;            // tile_dim1=128, tile_dim2=0
        g1[5] = 128u;            // tensor_dim0_stride[31:0]=128
        g1[6] = (16384u << 16);  // stride0[47:32]=0 | tensor_dim1_stride[15:0]
        g1[7] = 0u;              // tensor_dim1_stride[47:16]=0

        asm volatile("tensor_load_to_lds %0, %1" :: "s"(g0), "s"(g1) : "memory");
        __builtin_amdgcn_s_wait_tensorcnt(0);
    }
#else
    // ---- fallback: float4 staging through VGPRs ----
    #pragma unroll
    for (int it = 0; it < (GSZ * GSZ) / (256 * 4); ++it) {
        const int e   = (tid + 256 * it) * 4;
        const int row = e >> 7;
        const int col = e & (GSZ - 1);
        const float4 v = *(const float4*)(xg + e);
        float* d = &xs[row * PAD + col];
        d[0] = v.x; d[1] = v.y; d[2] = v.z; d[3] = v.w;
    }
#endif
    // r0 = ones
    if (tid < GSZ) rall[0][tid] = 1.0f;
    __syncthreads();

    const int wave    = tid >> 5;        // 0..7 -> 16-row block
    const int lane    = tid & 31;
    const int m       = lane & 15;
    const int hi      = lane >> 4;       // 0: K=0,1   1: K=2,3
    const int rowbase = wave * 16;

    // ---- matvec chain: rall[s+1] = X * rall[s], s = 0..3 ----
    for (int step = 0; step < NPOW; ++step) {
        const float* __restrict__ rsrc = rall[step];
        v8f c = {};   // 16x16 f32 accumulator (all 16 columns identical)

        #pragma unroll
        for (int cb = 0; cb < GSZ; cb += 4) {
            const int colA = cb + hi * 2;
            // A fragment: A[m][k] = X[rowbase+m][cb+k]
            v2f a = *(const v2f*)&xs[(rowbase + m) * PAD + colA];
            // B fragment broadcast over N: B[k][n] = r[cb+k] for all n
            v2f b;
            b.x = rsrc[colA];
            b.y = rsrc[colA + 1];
            c = __builtin_amdgcn_wmma_f32_16x16x4_f32(
                    /*neg_a=*/false, a, /*neg_b=*/false, b,
                    /*c_mod=*/(short)0, c, /*reuse_a=*/false, /*reuse_b=*/false);
        }

        // D layout: VGPR p, lanes 0-15 -> M=p; lanes 16-31 -> M=p+8 (all N equal)
        if (m == 0) {
            float* rdst = rall[step + 1];
            #pragma unroll
            for (int p = 0; p < 8; ++p)
                rdst[rowbase + hi * 8 + p] = c[p];
        }
        __syncthreads();
    }

    // ---- final scalar combine (tiny; hidden behind 4096-block parallelism) ----
    if (tid == 0) {
        float S[3];
        #pragma unroll
        for (int k = 0; k < 3; ++k) {
            float acc = 0.0f;
            for (int q = 0; q < GSZ; ++q) acc += rall[k + 2][q];  // sum(x^(k+2))
            S[k] = acc;
        }
        const float ninv = 1.0f / (float)(GSZ * GSZ);   // 2^-14, exact
        float ip[7];
        ip[0] = 1.0f;
        #pragma unroll
        for (int k = 1; k < 7; ++k) ip[k] = ip[k - 1] * ninv;

        float res = 0.0f;
        #pragma unroll
        for (int i = 0; i < 3; ++i) {
            float p = 1.0f;
            #pragma unroll
            for (int j = 0; j < 3; ++j) {
                p *= S[i];                               // S^(j+1)
                res += coef[i * 3 + j] * p * ip[i + j + 2];
            }
        }
        out[bid] = res;
    }
}

extern "C" void kernel_launch(void* const* d_in, const int* in_sizes, int n_in,
                              void* d_out, int out_size, void* d_ws, size_t ws_size,
                              hipStream_t stream)
{
    (void)n_in; (void)d_ws; (void)ws_size; (void)out_size;
    const float* x    = (const float*)d_in[0];
    const float* coef = (const float*)d_in[1];
    float* out        = (float*)d_out;
    const int batch   = in_sizes[0] / (GSZ * GSZ);   // 4096
    facoef_kernel<<<batch, 256, 0, stream>>>(x, coef, out);
}